// HashEmbedder_8177617732077
// MI455X (gfx1250) — compile-verified
//
#include <hip/hip_runtime.h>

// ---------------- Hash-grid constants (match reference) ----------------
#define N_LEVELS     16
#define LOG2_T       19
#define TABLE_SIZE   (1u << LOG2_T)
#define TMASK        (TABLE_SIZE - 1u)
#define PRIME_Y      2654435761u
#define PRIME_Z      805459861u

#define PTS_PER_BLOCK 32
#define BLOCK_THREADS 512   // 16 levels x 32 points; one wave32 == one level

typedef float v2f __attribute__((ext_vector_type(2)));   // native vector for NT builtins

// res[l] = floor(16 * (32^(1/15))^l) = floor(16 * 2^(l/3))
// -> 16,20,25,32,40,50,64,80,101,128,161,203,256,322,406,512
// grid = 6/res ; invg = 1/grid  (compile-time f32 rounding)
#define GRID_OF(R)  (6.0f / (R))
#define INVG_OF(R)  (1.0f / (6.0f / (R)))

__constant__ float c_grid[N_LEVELS] = {
    GRID_OF(16.0f),  GRID_OF(20.0f),  GRID_OF(25.0f),  GRID_OF(32.0f),
    GRID_OF(40.0f),  GRID_OF(50.0f),  GRID_OF(64.0f),  GRID_OF(80.0f),
    GRID_OF(101.0f), GRID_OF(128.0f), GRID_OF(161.0f), GRID_OF(203.0f),
    GRID_OF(256.0f), GRID_OF(322.0f), GRID_OF(406.0f), GRID_OF(512.0f)
};
__constant__ float c_invg[N_LEVELS] = {
    INVG_OF(16.0f),  INVG_OF(20.0f),  INVG_OF(25.0f),  INVG_OF(32.0f),
    INVG_OF(40.0f),  INVG_OF(50.0f),  INVG_OF(64.0f),  INVG_OF(80.0f),
    INVG_OF(101.0f), INVG_OF(128.0f), INVG_OF(161.0f), INVG_OF(203.0f),
    INVG_OF(256.0f), INVG_OF(322.0f), INVG_OF(406.0f), INVG_OF(512.0f)
};

__global__ void __launch_bounds__(BLOCK_THREADS)
hashgrid_encode_kernel(const float* __restrict__ xg,
                       const float* __restrict__ tables,
                       float* __restrict__ out,
                       int B)
{
    __shared__ float sX[PTS_PER_BLOCK * 3];        // staged coordinates
    __shared__ float sOut[PTS_PER_BLOCK * 33];     // 33-word pitch: conflict-free

    const int tid = threadIdx.x;
    const int p0  = blockIdx.x * PTS_PER_BLOCK;

    // ---- stage 32 points' coords (coalesced 384B, non-temporal: read once) ----
    if (tid < PTS_PER_BLOCK * 3) {
        const int gi = p0 * 3 + tid;               // fits in int32 (3M max)
        sX[tid] = (gi < B * 3) ? __builtin_nontemporal_load(xg + gi) : 0.0f;
    }
    __syncthreads();

    // wave32 layout: lanes = 32 points, each wave owns one level
    const int level = __builtin_amdgcn_readfirstlane(tid >> 5);   // wave-uniform
    const int ptl   = tid & 31;
    const int p     = p0 + ptl;

    const float grid = c_grid[level];
    const float invg = c_invg[level];
    const v2f* __restrict__ tbl =
        (const v2f*)tables + (size_t)level * TABLE_SIZE;

    const float px = sX[ptl * 3 + 0];
    const float py = sX[ptl * 3 + 1];
    const float pz = sX[ptl * 3 + 2];

    // clamp only for cell selection; weights use UNCLAMPED coords (as reference)
    const float cx = fminf(fmaxf(px, -3.0f), 3.0f);
    const float cy = fminf(fmaxf(py, -3.0f), 3.0f);
    const float cz = fminf(fmaxf(pz, -3.0f), 3.0f);

    const float blx = floorf((cx + 3.0f) * invg);
    const float bly = floorf((cy + 3.0f) * invg);
    const float blz = floorf((cz + 3.0f) * invg);

    const float wx1 = (px - (blx * grid - 3.0f)) * invg;
    const float wy1 = (py - (bly * grid - 3.0f)) * invg;
    const float wz1 = (pz - (blz * grid - 3.0f)) * invg;
    const float wx0 = 1.0f - wx1, wy0 = 1.0f - wy1, wz0 = 1.0f - wz1;

    // incremental spatial hash (uint32 wraparound identical to reference)
    const unsigned hx0 = (unsigned)(int)blx;                 // prime 1
    const unsigned hx1 = hx0 + 1u;
    const unsigned hy0 = (unsigned)(int)bly * PRIME_Y;
    const unsigned hy1 = hy0 + PRIME_Y;
    const unsigned hz0 = (unsigned)(int)blz * PRIME_Z;
    const unsigned hz1 = hz0 + PRIME_Z;

    // 8 gathers issued up front so loads overlap, then FMA chain (ref order: i-major).
    // Default (RT) temporal hint: tables must STAY resident in the 192MB L2.
    const v2f e000 = tbl[(hx0 ^ hy0 ^ hz0) & TMASK];
    const v2f e001 = tbl[(hx0 ^ hy0 ^ hz1) & TMASK];
    const v2f e010 = tbl[(hx0 ^ hy1 ^ hz0) & TMASK];
    const v2f e011 = tbl[(hx0 ^ hy1 ^ hz1) & TMASK];
    const v2f e100 = tbl[(hx1 ^ hy0 ^ hz0) & TMASK];
    const v2f e101 = tbl[(hx1 ^ hy0 ^ hz1) & TMASK];
    const v2f e110 = tbl[(hx1 ^ hy1 ^ hz0) & TMASK];
    const v2f e111 = tbl[(hx1 ^ hy1 ^ hz1) & TMASK];

    float a0 = 0.0f, a1 = 0.0f;
    {
        float w;
        w = wx0 * wy0 * wz0; a0 = fmaf(w, e000.x, a0); a1 = fmaf(w, e000.y, a1);
        w = wx0 * wy0 * wz1; a0 = fmaf(w, e001.x, a0); a1 = fmaf(w, e001.y, a1);
        w = wx0 * wy1 * wz0; a0 = fmaf(w, e010.x, a0); a1 = fmaf(w, e010.y, a1);
        w = wx0 * wy1 * wz1; a0 = fmaf(w, e011.x, a0); a1 = fmaf(w, e011.y, a1);
        w = wx1 * wy0 * wz0; a0 = fmaf(w, e100.x, a0); a1 = fmaf(w, e100.y, a1);
        w = wx1 * wy0 * wz1; a0 = fmaf(w, e101.x, a0); a1 = fmaf(w, e101.y, a1);
        w = wx1 * wy1 * wz0; a0 = fmaf(w, e110.x, a0); a1 = fmaf(w, e110.y, a1);
        w = wx1 * wy1 * wz1; a0 = fmaf(w, e111.x, a0); a1 = fmaf(w, e111.y, a1);
    }

    if (p < B) {
        sOut[ptl * 33 + level * 2 + 0] = a0;   // stride-33: conflict-free banks
        sOut[ptl * 33 + level * 2 + 1] = a1;
    }
    __syncthreads();

    // ---- coalesced write-out: lane-consecutive 8B NT stores ----
    // Output is 128MB written once and never re-read: non-temporal keeps it
    // from evicting the 64MB of gather tables out of L2 (64+128 ~= L2 size).
    const int pt  = tid >> 4;            // 0..31
    const int rem = (tid * 2) & 31;      // even, 0..30  (level*2 + feat)
    const int gp  = p0 + pt;
    if (gp < B) {
        v2f v;
        v.x = sOut[pt * 33 + rem];
        v.y = sOut[pt * 33 + rem + 1];
        __builtin_nontemporal_store(v, (v2f*)(out + (size_t)gp * 32 + rem));
    }
}

extern "C" void kernel_launch(void* const* d_in, const int* in_sizes, int n_in,
                              void* d_out, int out_size, void* d_ws, size_t ws_size,
                              hipStream_t stream) {
    const float* x      = (const float*)d_in[0];   // [B,3] f32
    const float* tables = (const float*)d_in[1];   // [16, 2^19, 2] f32
    float*       out    = (float*)d_out;           // [B,32] f32

    const int B = in_sizes[0] / 3;
    const int nblocks = (B + PTS_PER_BLOCK - 1) / PTS_PER_BLOCK;

    hipLaunchKernelGGL(hashgrid_encode_kernel,
                       dim3(nblocks), dim3(BLOCK_THREADS), 0, stream,
                       x, tables, out, B);
}